// KPConv_63170378990247
// MI455X (gfx1250) — compile-verified
//
#include <hip/hip_runtime.h>

typedef float v2f __attribute__((ext_vector_type(2)));
typedef float v8f __attribute__((ext_vector_type(8)));

#define NQ_TILE 16
#define K_NEIGH 32
#define KP      15
#define CIN     64
#define COUT    64
#define KDIM    (KP * CIN)   // 960 = flattened (p,c) reduction dim
#define ROW     964          // LDS row stride (960 + 4): 16 rows -> 16 distinct banks

__global__ __launch_bounds__(128) void kpconv_fused_wmma(
    const float* __restrict__ s_feats,    // [Ns, 64]
    const float* __restrict__ q_points,   // [Nq, 3]
    const float* __restrict__ s_points,   // [Ns, 3]
    const int*   __restrict__ nb_idx,     // [Nq, 32]
    const float* __restrict__ kpts,       // [15, 3]
    const float* __restrict__ weights,    // [15, 64, 64] == [960, 64]
    float* __restrict__ out,              // [Nq, 64]
    int Nq)
{
    __shared__ float smem[NQ_TILE * ROW];   // weighted[16][960] (padded)
    __shared__ float sInv[NQ_TILE];         // 1 / neighbor_num

    const int tid  = threadIdx.x;
    const int lane = tid & 31;
    const int wave = tid >> 5;
    const int n0   = blockIdx.x * NQ_TILE;

    // kernel point dispositions in registers (45 floats, broadcast loads)
    float kpx[KP], kpy[KP], kpz[KP];
#pragma unroll
    for (int p = 0; p < KP; ++p) {
        kpx[p] = kpts[p * 3 + 0];
        kpy[p] = kpts[p * 3 + 1];
        kpz[p] = kpts[p * 3 + 2];
    }

    // ---------------- Phase 1: build weighted[16][960] in LDS ----------------
    // wave handles queries r = wave*4 .. wave*4+3 ; lane owns channels lane, lane+32
    for (int i = 0; i < 4; ++i) {
        const int r = wave * 4 + i;
        const int n = n0 + r;

        const float qx = q_points[n * 3 + 0];
        const float qy = q_points[n * 3 + 1];
        const float qz = q_points[n * 3 + 2];
        const int myidx = nb_idx[n * K_NEIGH + lane];   // coalesced, one per lane

        float acc0[KP], acc1[KP];
#pragma unroll
        for (int p = 0; p < KP; ++p) { acc0[p] = 0.0f; acc1[p] = 0.0f; }
        int cnt = 0;

        for (int k = 0; k < K_NEIGH; ++k) {
            const int idx = __shfl(myidx, k, 32);       // broadcast neighbor k
            const float sx = s_points[idx * 3 + 0] - qx;
            const float sy = s_points[idx * 3 + 1] - qy;
            const float sz = s_points[idx * 3 + 2] - qz;

            const float x0 = s_feats[idx * CIN + lane];        // channels 0..31
            const float x1 = s_feats[idx * CIN + 32 + lane];   // channels 32..63

            // neighbor feature sum over all 64 channels (valid-neighbor count)
            float s = x0 + x1;
#pragma unroll
            for (int off = 16; off > 0; off >>= 1) s += __shfl_xor(s, off, 32);
            cnt += (s > 0.0f) ? 1 : 0;   // uniform across wave

#pragma unroll
            for (int p = 0; p < KP; ++p) {
                const float dx = sx - kpx[p];
                const float dy = sy - kpy[p];
                const float dz = sz - kpz[p];
                const float d2 = dx * dx + dy * dy + dz * dz;
                // SIGMA = 2.0 -> 1/SIGMA = 0.5
                const float w = fmaxf(1.0f - sqrtf(d2) * 0.5f, 0.0f);
                acc0[p] = fmaf(w, x0, acc0[p]);
                acc1[p] = fmaf(w, x1, acc1[p]);
            }
        }

#pragma unroll
        for (int p = 0; p < KP; ++p) {
            smem[r * ROW + p * CIN + lane]      = acc0[p];
            smem[r * ROW + p * CIN + 32 + lane] = acc1[p];
        }
        if (lane == 0) {
            const int nn = (cnt < 1) ? 1 : cnt;
            sInv[r] = 1.0f / (float)nn;
        }
    }
    __syncthreads();

    // ---------------- Phase 2: 16x64 GEMM tile via V_WMMA_F32_16X16X4_F32 ----
    // wave w owns output columns [w*16, w*16+16)
    const int col  = wave * 16 + (lane & 15);
    const int arow = lane & 15;              // A-matrix row = query row in tile
    const int koff = (lane < 16) ? 0 : 2;    // lanes 0-15: K0,K1 ; lanes 16-31: K2,K3

    v8f acc = {0.f, 0.f, 0.f, 0.f, 0.f, 0.f, 0.f, 0.f};

#pragma unroll 4
    for (int k = 0; k < KDIM; k += 4) {
        // A 16x4 fragment from LDS (row-major weighted, padded stride)
        v2f a = *(const v2f*)&smem[arow * ROW + k + koff];
        // B 4x16 fragment from global weights (L2-resident, reused by all blocks)
        v2f b;
        b.x = weights[(k + koff)     * COUT + col];
        b.y = weights[(k + koff + 1) * COUT + col];
        acc = __builtin_amdgcn_wmma_f32_16x16x4_f32(false, a, false, b,
                                                    (short)0, acc, false, false);
    }

    // Epilogue: density normalization + store.
    // C/D layout: VGPR j holds row M=j (lanes 0-15) and M=j+8 (lanes 16-31)
#pragma unroll
    for (int j = 0; j < 8; ++j) {
        const int m = j + ((lane >> 4) << 3);
        out[(n0 + m) * COUT + col] = acc[j] * sInv[m];
    }
}

extern "C" void kernel_launch(void* const* d_in, const int* in_sizes, int n_in,
                              void* d_out, int out_size, void* d_ws, size_t ws_size,
                              hipStream_t stream) {
    const float* s_feats  = (const float*)d_in[0];
    const float* q_points = (const float*)d_in[1];
    const float* s_points = (const float*)d_in[2];
    const int*   nb_idx   = (const int*)  d_in[3];
    const float* kpts     = (const float*)d_in[4];
    const float* weights  = (const float*)d_in[5];
    float* out = (float*)d_out;

    const int Nq = in_sizes[1] / 3;   // q_points is [Nq, 3]
    const int grid = Nq / NQ_TILE;    // 50000 / 16 = 3125, exact

    kpconv_fused_wmma<<<grid, 128, 0, stream>>>(
        s_feats, q_points, s_points, nb_idx, kpts, weights, out, Nq);
}